// DarcyFlow_88304527606507
// MI455X (gfx1250) — compile-verified
//
#include <hip/hip_runtime.h>
#include <stdint.h>

// ---------------------------------------------------------------------------
// DarcyFlow stencil for MI455X (gfx1250).
// Ax[i,j] = sum over 4 diagonal dirs (dh,dw in {-1,+1}) of
//   a[i+(dh-1)/2, j+(dw-1)/2] * ( (2/3)x[i,j] - (1/6)x[i+dh,j]
//                                 - (1/6)x[i,j+dw] - (1/3)x[i+dh,j+dw] )
// for i,j in [1,1022]; zero border. Memory-bound (~2.8 FLOP/B) -> TDM async
// tile loads to LDS + vectorized f32 VALU compute.
// ---------------------------------------------------------------------------

#define NN   1024
#define NAA  1023
#define TW   128
#define TH   16
#define XTW  (TW + 2)   // 130
#define XTH  (TH + 2)   // 18
#define ATW  (TW + 1)   // 129
#define ATH  (TH + 1)   // 17

typedef unsigned int v4u __attribute__((ext_vector_type(4)));
typedef int          v4i __attribute__((ext_vector_type(4)));
typedef int          v8i __attribute__((ext_vector_type(8)));

#if defined(__HIP_DEVICE_COMPILE__) && __has_builtin(__builtin_amdgcn_tensor_load_to_lds)
#define HAVE_TDM 1
#else
#define HAVE_TDM 0
#endif

__device__ __forceinline__ int imin(int a, int b) { return a < b ? a : b; }
__device__ __forceinline__ int imax(int a, int b) { return a > b ? a : b; }

#if HAVE_TDM
// 2D tile DMA: global (tdim1 x tdim0 tensor, row pitch stride0 elements,
// 4-byte elements) -> LDS, rows packed contiguously (LDS row stride = tile0).
__device__ __forceinline__ void tdm_load_2d(uint32_t lds_byte_addr,
                                            const void* gptr,
                                            uint32_t tdim0, uint32_t tdim1,
                                            uint32_t tile0, uint32_t tile1,
                                            uint32_t stride0) {
  uint64_t ga = (uint64_t)(uintptr_t)gptr;
  v4u g0;
  g0.x = 1u;                                                   // count=1 (valid)
  g0.y = lds_byte_addr;                                        // lds_addr
  g0.z = (uint32_t)ga;                                         // global_addr lo
  g0.w = ((uint32_t)(ga >> 32) & 0x01FFFFFFu) | 0x80000000u;   // addr hi | type=2
  v8i g1;
  g1[0] = (int)(2u << 16);                                     // data_size=4B
  g1[1] = (int)((tdim0 & 0xFFFFu) << 16);                      // tensor_dim0[15:0]
  g1[2] = (int)((tdim0 >> 16) | ((tdim1 & 0xFFFFu) << 16));    // dim0 hi | dim1 lo
  g1[3] = (int)((tdim1 >> 16) | (tile0 << 16));                // dim1 hi | tile_dim0
  g1[4] = (int)(tile1 & 0xFFFFu);                              // tile_dim1, tile_dim2=0
  g1[5] = (int)stride0;                                        // tensor_dim0_stride lo
  g1[6] = 0;
  g1[7] = 0;
  v4i g2 = {1, 0, 0, 0};            // tensor_dim2 = 1 (unused dims benign)
  v4i g3 = {0, 1 << 16, 0, 0};      // tensor_dim4 = 1
#if defined(__clang_major__) && (__clang_major__ >= 23)
  v8i z8 = {0, 0, 0, 0, 0, 0, 0, 0};
  __builtin_amdgcn_tensor_load_to_lds(g0, g1, g2, g3, z8, 0);
#else
  __builtin_amdgcn_tensor_load_to_lds(g0, g1, g2, g3, 0);
#endif
}
#endif

__global__ __launch_bounds__(256) void darcy_stencil_kernel(
    const float* __restrict__ x, const float* __restrict__ a,
    float* __restrict__ out) {
  __shared__ float xs[XTH * XTW];   // 2340 floats
  __shared__ float as_[ATH * ATW];  // 2193 floats

  const int c0 = blockIdx.x * TW;   // first output col of tile
  const int r0 = blockIdx.y * TH;   // first output row of tile
  const int b  = blockIdx.z;

  // Clamped halo tile origins / extents (all loads stay in-bounds).
  const int cx0 = imax(c0 - 1, 0);
  const int rx0 = imax(r0 - 1, 0);
  const int xw  = imin(c0 + TW, NN - 1) - cx0 + 1;        // <= 130
  const int xh  = imin(r0 + TH, NN - 1) - rx0 + 1;        // <= 18
  const int aw  = imin(c0 + TW - 1, NAA - 1) - cx0 + 1;   // <= 129
  const int ah  = imin(r0 + TH - 1, NAA - 1) - rx0 + 1;   // <= 17
  const int bxc = c0 - cx0;  // 0 at left edge, else 1
  const int bxr = r0 - rx0;  // 0 at top edge,  else 1

#if HAVE_TDM
  {
    const unsigned wid = (unsigned)__builtin_amdgcn_readfirstlane((int)threadIdx.x) >> 5;
    if (wid == 0) {
      const uint32_t xs_lds = (uint32_t)(uintptr_t)(&xs[0]);
      const uint32_t as_lds = (uint32_t)(uintptr_t)(&as_[0]);
      const float* gx = x + ((size_t)b * NN * NN + (size_t)rx0 * NN + cx0);
      tdm_load_2d(xs_lds, gx, (uint32_t)(NN - cx0), (uint32_t)(NN - rx0),
                  (uint32_t)xw, (uint32_t)xh, (uint32_t)NN);
      const float* ga = a + ((size_t)rx0 * NAA + cx0);
      tdm_load_2d(as_lds, ga, (uint32_t)(NAA - cx0), (uint32_t)(NAA - rx0),
                  (uint32_t)aw, (uint32_t)ah, (uint32_t)NAA);
      __builtin_amdgcn_s_wait_tensorcnt(0);
    }
  }
#else
  for (int t = threadIdx.x; t < xh * xw; t += blockDim.x) {
    const int rr = t / xw, cc = t - rr * xw;
    xs[t] = x[(size_t)b * NN * NN + (size_t)(rx0 + rr) * NN + (cx0 + cc)];
  }
  for (int t = threadIdx.x; t < ah * aw; t += blockDim.x) {
    const int rr = t / aw, cc = t - rr * aw;
    as_[t] = a[(size_t)(rx0 + rr) * NAA + (cx0 + cc)];
  }
#endif
  __syncthreads();

  const int tx = threadIdx.x & 31;   // 32 col-groups of 4
  const int ty = threadIdx.x >> 5;   // 8 row slots, 2 rows each
  const int lj = tx * 4;

  const float C23 = 2.0f / 3.0f, C16 = 1.0f / 6.0f, C13 = 1.0f / 3.0f;

#pragma unroll
  for (int rr = 0; rr < TH; rr += 8) {
    const int li  = ty + rr;
    const int gi  = r0 + li;
    const int gj0 = c0 + lj;

    // LDS pointers: x rows (gi-1,gi,gi+1), cols starting at (gj0-1);
    // a rows (gi-1,gi), cols starting at (gj0-1). Runtime stride = loaded width.
    const int xbase = (li + bxr - 1) * xw + (lj + bxc - 1);
    const float* x0 = &xs[xbase];
    const float* x1 = x0 + xw;
    const float* x2 = x1 + xw;
    const int abase = (li + bxr - 1) * aw + (lj + bxc - 1);
    const float* a0 = &as_[abase];
    const float* a1 = a0 + aw;

    float o[4];
#pragma unroll
    for (int k = 0; k < 4; ++k) {
      const float xul = x0[k], xu = x0[k + 1], xur = x0[k + 2];
      const float xl  = x1[k], cc = x1[k + 1], xr_ = x1[k + 2];
      const float xdl = x2[k], xd = x2[k + 1], xdr = x2[k + 2];
      const float t0 = C23 * cc - C16 * xu - C16 * xl  - C13 * xul;  // (-1,-1)
      const float t1 = C23 * cc - C16 * xu - C16 * xr_ - C13 * xur;  // (-1,+1)
      const float t2 = C23 * cc - C16 * xd - C16 * xl  - C13 * xdl;  // (+1,-1)
      const float t3 = C23 * cc - C16 * xd - C16 * xr_ - C13 * xdr;  // (+1,+1)
      const float v  = a0[k] * t0 + a0[k + 1] * t1 + a1[k] * t2 + a1[k + 1] * t3;
      const int gj = gj0 + k;
      const bool valid = (gi >= 1) && (gi <= NN - 2) && (gj >= 1) && (gj <= NN - 2);
      o[k] = valid ? v : 0.0f;   // border (and masked lanes) write zeros
    }

    // 16B-aligned vector store; covers every element of out exactly once.
    float4* op = (float4*)(out + (size_t)b * NN * NN + (size_t)gi * NN + gj0);
    *op = make_float4(o[0], o[1], o[2], o[3]);
  }
}

extern "C" void kernel_launch(void* const* d_in, const int* in_sizes, int n_in,
                              void* d_out, int out_size, void* d_ws, size_t ws_size,
                              hipStream_t stream) {
  (void)n_in; (void)out_size; (void)d_ws; (void)ws_size;
  const float* x = (const float*)d_in[0];
  const float* a = (const float*)d_in[1];
  float* out = (float*)d_out;
  const int B = in_sizes[0] / (NN * NN);   // 16

  dim3 grid(NN / TW, NN / TH, B);          // 8 x 64 x 16
  dim3 block(256);
  darcy_stencil_kernel<<<grid, block, 0, stream>>>(x, a, out);
}